// MomentCalculator_80401787781603
// MI455X (gfx1250) — compile-verified
//
#include <hip/hip_runtime.h>
#include <math.h>

#define AS1 __attribute__((address_space(1)))
#define AS3 __attribute__((address_space(3)))

typedef __attribute__((ext_vector_type(2))) float v2f;
typedef __attribute__((ext_vector_type(8))) float v8f;
typedef __attribute__((ext_vector_type(4))) int   v4i;

#define ROW_LEN 16384
#define WAVE_SLICE 2048   // ROW_LEN / 8 waves
#define NPART 80          // 8 mean partials + 8 waves * 9 moment partials

// Accumulate row-sums of a 16x4 f32 tile (A) into C via the matrix pipe.
// B is all-ones so C[m][n] += sum_k A[m][k]; total over C == total over A,
// independent of the lane->(m,k) mapping.
#define WMMA_SUM(C, A) \
  (C) = __builtin_amdgcn_wmma_f32_16x16x4_f32(false, (A), false, ones, (short)0, (C), false, false)

// Full-wave total of a 16x16 f32 C tile: columns are identical; VGPRs 0..7
// hold rows 0..7 (lanes 0-15) / rows 8..15 (lanes 16-31).
static __device__ __forceinline__ float wave_total(v8f c) {
  float t = c[0] + c[1] + c[2] + c[3] + c[4] + c[5] + c[6] + c[7];
  t += __shfl_xor(t, 16, 32);   // wave32: xor lane 16 merges the two row halves
  return t;
}

static __device__ __forceinline__ float sgnlog(float x) {
  float s = (x > 0.0f) ? 1.0f : ((x < 0.0f) ? -1.0f : 0.0f);
  return s * logf(fabsf(x) + 1e-10f);
}

__global__ void MomentCalculator_kernel(const float* __restrict__ y,
                                        float* __restrict__ out, int B) {
  extern __shared__ float smem[];
  float* data = smem;            // ROW_LEN floats: staged row
  float* part = smem + ROW_LEN;  // NPART floats: cross-wave partials

  const int tid  = threadIdx.x;
  const int lane = tid & 31;
  const int wave = tid >> 5;
  const int row  = blockIdx.x;

  const float* grow = y + (size_t)row * ROW_LEN;

  // ---- Stage the 64KB row into LDS (async global->LDS, ASYNCcnt path) ----
#if defined(__gfx1250__) && __has_builtin(__builtin_amdgcn_global_load_async_to_lds_b128)
  {
    float* gmut = const_cast<float*>(grow);
#pragma unroll
    for (int i = 0; i < 16; ++i) {
      const int e = (i * 256 + tid) * 4;  // 16B per thread-chunk, coalesced
      __builtin_amdgcn_global_load_async_to_lds_b128(
          (AS1 v4i*)(gmut + e), (AS3 v4i*)(data + e), 0, 0);
    }
#if __has_builtin(__builtin_amdgcn_s_wait_asynccnt)
    __builtin_amdgcn_s_wait_asynccnt(0);
#else
    asm volatile("s_wait_asynccnt 0" ::: "memory");
#endif
  }
#else
  {
    const float4* g4 = (const float4*)grow;
    float4* l4 = (float4*)data;
#pragma unroll
    for (int i = 0; i < 16; ++i) {
      const int e = i * 256 + tid;
      l4[e] = g4[e];
    }
  }
#endif
  __syncthreads();

  const v2f ones = {1.0f, 1.0f};
  const int wbase = wave * WAVE_SLICE;

  // ---- Pass 1: row sum -> mean (WMMA ones-reduction) ----
  v8f cs = {};
#pragma unroll 4
  for (int it = 0; it < 32; ++it) {
    v2f a = *(const v2f*)(data + wbase + it * 64 + lane * 2);
    WMMA_SUM(cs, a);
  }
  const float wsum = wave_total(cs);
  if (lane == 0) part[wave] = wsum;
  __syncthreads();

  float total = 0.0f;
#pragma unroll
  for (int w = 0; w < 8; ++w) total += part[w];
  const float mean = total * (1.0f / (float)ROW_LEN);

  // ---- Pass 2: central moment sums z^2..z^10, 9 WMMA accumulators ----
  v8f c2 = {}, c3 = {}, c4 = {}, c5 = {}, c6 = {}, c7 = {}, c8 = {}, c9 = {}, c10 = {};
#pragma unroll 2
  for (int it = 0; it < 32; ++it) {
    v2f yv = *(const v2f*)(data + wbase + it * 64 + lane * 2);
    v2f z = yv - mean;
    v2f zp = z;
    zp *= z; WMMA_SUM(c2, zp);
    zp *= z; WMMA_SUM(c3, zp);
    zp *= z; WMMA_SUM(c4, zp);
    zp *= z; WMMA_SUM(c5, zp);
    zp *= z; WMMA_SUM(c6, zp);
    zp *= z; WMMA_SUM(c7, zp);
    zp *= z; WMMA_SUM(c8, zp);
    zp *= z; WMMA_SUM(c9, zp);
    zp *= z; WMMA_SUM(c10, zp);
  }

  float m[9];
  m[0] = wave_total(c2);  m[1] = wave_total(c3);  m[2] = wave_total(c4);
  m[3] = wave_total(c5);  m[4] = wave_total(c6);  m[5] = wave_total(c7);
  m[6] = wave_total(c8);  m[7] = wave_total(c9);  m[8] = wave_total(c10);
  if (lane == 0) {
#pragma unroll
    for (int p = 0; p < 9; ++p) part[8 + wave * 9 + p] = m[p];
  }
  __syncthreads();

  // ---- Finalize: 10 scalars per row ----
  if (tid == 0) {
    const float EPS = 1e-10f;
    float S[9];
#pragma unroll
    for (int p = 0; p < 9; ++p) {
      float s = 0.0f;
      for (int w = 0; w < 8; ++w) s += part[8 + w * 9 + p];
      S[p] = s;
    }
    const float stdv = sqrtf(S[0] / (float)(ROW_LEN - 1)) + EPS;  // unbiased
    float* orow = out + (size_t)row * 10;
    orow[0] = sgnlog(mean);  // m0: not clamped, per reference
    float stdp = stdv;
#pragma unroll
    for (int p = 0; p < 9; ++p) {
      stdp *= stdv;  // stdv^(p+2)
      const float cm = S[p] * (1.0f / (float)ROW_LEN);
      const float nrm = cm / (stdp + EPS);
      const float v = sgnlog(nrm);
      orow[1 + p] = fminf(10.0f, fmaxf(-10.0f, v));
    }
  }
}

extern "C" void kernel_launch(void* const* d_in, const int* in_sizes, int n_in,
                              void* d_out, int out_size, void* d_ws, size_t ws_size,
                              hipStream_t stream) {
  (void)n_in; (void)d_ws; (void)ws_size; (void)out_size;
  const float* y = (const float*)d_in[0];
  float* out = (float*)d_out;
  const int B = in_sizes[0] / ROW_LEN;  // 8192 rows of 16384

  const size_t shmem = (size_t)(ROW_LEN + NPART) * sizeof(float);  // 65,856 B
  // LDS/WG can be up to 320KB on CDNA5; raise the runtime cap above 64KB.
  // (Attribute set, not a stream op: deterministic and graph-capture safe.)
  (void)hipFuncSetAttribute((const void*)MomentCalculator_kernel,
                            hipFuncAttributeMaxDynamicSharedMemorySize,
                            (int)shmem);

  hipLaunchKernelGGL(MomentCalculator_kernel, dim3(B), dim3(256), shmem, stream,
                     y, out, B);
}